// SplineBlock_73684458930803
// MI455X (gfx1250) — compile-verified
//
#include <hip/hip_runtime.h>
#include <hip/hip_bf16.h>

// Rational-quadratic spline evaluation, N=4e6 rows, K=8 knots.
// Pure streaming kernel: 104 B/row @ 23.3 TB/s => ~18 us floor.
// One row per lane, b128 non-temporal loads, branch-free select-scan gather.

typedef __attribute__((ext_vector_type(4))) float v4f;

__global__ __launch_bounds__(256) void spline_rq_kernel(
    const float* __restrict__ x,
    const float* __restrict__ t,
    const float* __restrict__ y,
    const float* __restrict__ d,
    float* __restrict__ out,
    int n)
{
    int r = blockIdx.x * blockDim.x + threadIdx.x;
    if (r >= n) return;

    // x is (N,1): coalesced b32 load per lane.
    float xs = __builtin_nontemporal_load(x + r);

    // Rows are 8 floats = 32 B, 32-B aligned -> two global_load_b128 each,
    // non-temporal (stream-once working set, 416 MB > 192 MB L2).
    const v4f* tp = (const v4f*)(t + (size_t)r * 8);
    const v4f* yp = (const v4f*)(y + (size_t)r * 8);
    const v4f* dp = (const v4f*)(d + (size_t)r * 8);
    v4f tl = __builtin_nontemporal_load(tp);
    v4f th = __builtin_nontemporal_load(tp + 1);
    v4f yl = __builtin_nontemporal_load(yp);
    v4f yh = __builtin_nontemporal_load(yp + 1);
    v4f dl = __builtin_nontemporal_load(dp);
    v4f dh = __builtin_nontemporal_load(dp + 1);

    float tv[8] = {tl.x, tl.y, tl.z, tl.w, th.x, th.y, th.z, th.w};
    float yv[8] = {yl.x, yl.y, yl.z, yl.w, yh.x, yh.y, yh.z, yh.w};
    float dv[8] = {dl.x, dl.y, dl.z, dl.w, dh.x, dh.y, dh.z, dh.w};

    // Bucket search fused with gather.
    // i = #{k : xs > t[k]}, k0 = clip(i-1, 0, 6), k1 = k0+1.
    // Since t is monotone: (xs > t[k]) <=> i-1 >= k. So scanning k=1..6 and
    // latching values[k] whenever (xs > t[k]) yields exactly index min(i-1,6)
    // (default 0 covers i<=1). Branch-free: pure v_cmp + v_cndmask.
    float t0 = tv[0], t1 = tv[1];
    float y0 = yv[0], y1 = yv[1];
    float d0 = dv[0], d1 = dv[1];
#pragma unroll
    for (int k = 1; k <= 6; ++k) {
        bool c = xs > tv[k];
        t0 = c ? tv[k]     : t0;
        t1 = c ? tv[k + 1] : t1;
        y0 = c ? yv[k]     : y0;
        y1 = c ? yv[k + 1] : y1;
        d0 = c ? dv[k]     : d0;
        d1 = c ? dv[k + 1] : d1;
    }
    bool eq0 = !(xs > tv[0]);   // i == 0
    bool eq1 =  (xs > tv[7]);   // i == 8

    // Rational-quadratic interior segment.
    float delta_t = t1 - t0;
    float delta_y = y1 - y0;
    float inv_dt  = 1.0f / delta_t;
    float s       = delta_y * inv_dt;
    float e       = (xs - t0) * inv_dt;
    float eome    = e * (1.0f - e);
    float n0      = delta_y * fmaf(s * e, e, d0 * eome);
    float n1      = fmaf(fmaf(-2.0f, s, d1 + d0), eome, s);
    float p_int   = y0 + n0 / n1;

    // Linear extrapolation tails.
    float p_lo = fmaf(d0, xs, fmaf(-d0, t0, y0));
    float p_hi = fmaf(d1, xs, fmaf(-d1, t1, y1));

    float p = eq0 ? p_lo : (eq1 ? p_hi : p_int);
    __builtin_nontemporal_store(p, out + r);
}

extern "C" void kernel_launch(void* const* d_in, const int* in_sizes, int n_in,
                              void* d_out, int out_size, void* d_ws, size_t ws_size,
                              hipStream_t stream) {
    // setup_inputs() dict order: x (N,1), t (N,8), y (N,8), d (N,8); all float32.
    const float* x = (const float*)d_in[0];
    const float* t = (const float*)d_in[1];
    const float* y = (const float*)d_in[2];
    const float* d = (const float*)d_in[3];
    float* out = (float*)d_out;

    int n = in_sizes[0];               // N rows (x is N elements)
    int threads = 256;                 // 8 wave32 per block
    int blocks = (n + threads - 1) / threads;
    spline_rq_kernel<<<blocks, threads, 0, stream>>>(x, t, y, d, out, n);
}